// HG_60481729462790
// MI455X (gfx1250) — compile-verified
//
#include <hip/hip_runtime.h>
#include <hip/hip_bf16.h>

typedef float v2f __attribute__((ext_vector_type(2)));
typedef float v8f __attribute__((ext_vector_type(8)));

#define HIDDEN 128

// ---------------------------------------------------------------------------
// Kernel 1: row L2-normalize (with +0.1 on the norm).  One wave32 per row.
// ---------------------------------------------------------------------------
__global__ __launch_bounds__(256) void hg_normalize(const float* __restrict__ emb,
                                                    float* __restrict__ out,
                                                    int nrows) {
    const int wave = (int)((blockIdx.x * blockDim.x + threadIdx.x) >> 5);
    const int lane = threadIdx.x & 31;
    if (wave >= nrows) return;

    const float4 v = ((const float4*)(emb + (size_t)wave * HIDDEN))[lane];
    float s = v.x * v.x + v.y * v.y + v.z * v.z + v.w * v.w;
    #pragma unroll
    for (int off = 16; off > 0; off >>= 1)
        s += __shfl_xor(s, off, 32);
    const float inv = 1.0f / (sqrtf(s) + 0.1f);
    float4 o;
    o.x = v.x * inv; o.y = v.y * inv; o.z = v.z * inv; o.w = v.w * inv;
    ((float4*)(out + (size_t)wave * HIDDEN))[lane] = o;
}

// ---------------------------------------------------------------------------
// Kernel 2: rna_t = rna_norm @ We  via V_WMMA_F32_16X16X4_F32.
// One wave per 16x16 tile of C; 8 waves/block cover the 8 N-tiles (N=128).
// A 16x4 f32 layout (ISA 7.12.2): lanes 0-15 hold M=0..15 / K=kp..kp+1 in
// 2 VGPRs, lanes 16-31 hold K=kp+2..kp+3.  B (4x16) row-striped across lanes.
// C: VGPR i -> row i (lanes 0-15) / row i+8 (lanes 16-31), col = lane&15.
// ---------------------------------------------------------------------------
__global__ __launch_bounds__(256) void hg_gemm(const float* __restrict__ An,
                                               const float* __restrict__ W,
                                               float* __restrict__ Ct,
                                               const int* __restrict__ d_rna_num) {
    const int rna_num = *d_rna_num;
    const int mtiles  = (rna_num + 15) >> 4;

    const int lane  = threadIdx.x & 31;
    const int wave  = threadIdx.x >> 5;      // 0..7 -> N tile
    const int n0    = wave << 4;
    const int mr    = lane & 15;
    const int kp    = (lane >> 4) << 1;      // 0 or 2 (upper half-wave)

    for (int mt = blockIdx.x; mt < mtiles; mt += gridDim.x) {
        const int m0 = mt << 4;
        int m  = m0 + mr;
        int mc = (m < rna_num) ? m : (rna_num - 1);   // clamp loads, guard stores
        const float* arow = An + (size_t)mc * HIDDEN;

        v8f c = {0.f, 0.f, 0.f, 0.f, 0.f, 0.f, 0.f, 0.f};
        #pragma unroll
        for (int k0 = 0; k0 < HIDDEN; k0 += 4) {
            v2f a, b;
            a.x = arow[k0 + kp];
            a.y = arow[k0 + kp + 1];
            b.x = W[(size_t)(k0 + kp)     * HIDDEN + n0 + mr];
            b.y = W[(size_t)(k0 + kp + 1) * HIDDEN + n0 + mr];
            c = __builtin_amdgcn_wmma_f32_16x16x4_f32(
                    /*neg_a=*/false, a, /*neg_b=*/false, b,
                    /*c_mod=*/(short)0, c, /*reuse_a=*/false, /*reuse_b=*/false);
        }

        const int row_half = (lane >> 4) << 3;      // 0 or 8
        const int col      = n0 + mr;
        #pragma unroll
        for (int i = 0; i < 8; ++i) {
            const int row = m0 + row_half + i;
            if (row < rna_num)
                Ct[(size_t)row * HIDDEN + col] = c[i];
        }
    }
}

// ---------------------------------------------------------------------------
// Kernel 3: per-edge dot + label.  One wave32 per edge; each lane loads a
// float4 slice of each 128-float row (coalesced 512B reads), shuffle-reduce.
// ---------------------------------------------------------------------------
__global__ __launch_bounds__(256) void hg_edges(const float* __restrict__ rt,
                                                const float* __restrict__ emb_norm,
                                                const int* __restrict__ d_rna_num,
                                                const int* __restrict__ pos_rna,
                                                const int* __restrict__ pos_dis,
                                                const int* __restrict__ neg_rna,
                                                const int* __restrict__ neg_dis,
                                                float* __restrict__ out,
                                                int epos, int eneg) {
    const int gw   = (int)((blockIdx.x * blockDim.x + threadIdx.x) >> 5);
    const int lane = threadIdx.x & 31;
    const int etot = epos + eneg;
    if (gw >= etot) return;

    const int rna_num = *d_rna_num;
    const float* dn = emb_norm + (size_t)rna_num * HIDDEN;

    int ri, di; float lab;
    if (gw < epos) { ri = pos_rna[gw];        di = pos_dis[gw];        lab = 1.0f; }
    else           { ri = neg_rna[gw - epos]; di = neg_dis[gw - epos]; lab = 0.0f; }

    const float4 a = ((const float4*)(rt + (size_t)ri * HIDDEN))[lane];
    const float4 b = ((const float4*)(dn + (size_t)di * HIDDEN))[lane];
    float s = a.x * b.x + a.y * b.y + a.z * b.z + a.w * b.w;
    #pragma unroll
    for (int off = 16; off > 0; off >>= 1)
        s += __shfl_xor(s, off, 32);

    if (lane == 0) {
        out[gw]        = s;     // logits (pos then neg)
        out[etot + gw] = lab;   // labels
    }
}

// ---------------------------------------------------------------------------
extern "C" void kernel_launch(void* const* d_in, const int* in_sizes, int n_in,
                              void* d_out, int out_size, void* d_ws, size_t ws_size,
                              hipStream_t stream) {
    const float* emb     = (const float*)d_in[0];
    const float* We      = (const float*)d_in[1];
    const int*   pos_rna = (const int*)d_in[2];
    const int*   pos_dis = (const int*)d_in[3];
    const int*   neg_rna = (const int*)d_in[4];
    const int*   neg_dis = (const int*)d_in[5];
    const int*   d_rnan  = (const int*)d_in[6];   // rna_num scalar on device
    float*       out     = (float*)d_out;

    const int total_rows = in_sizes[0] / HIDDEN;  // 25000
    const int epos       = in_sizes[2];           // 400000
    const int eneg       = in_sizes[4];           // 400000

    float* ws_norm = (float*)d_ws;                              // total_rows x 128
    float* ws_rt   = ws_norm + (size_t)total_rows * HIDDEN;     // rna rows x 128

    // 1) normalize all embedding rows (8 waves / block)
    const int nblocks = (total_rows + 7) / 8;
    hipLaunchKernelGGL(hg_normalize, dim3(nblocks), dim3(256), 0, stream,
                       emb, ws_norm, total_rows);

    // 2) rna_t = rna_norm @ We  (grid-stride over M tiles; rna_num read on device)
    const int mtiles_max = (total_rows + 15) / 16;
    hipLaunchKernelGGL(hg_gemm, dim3(mtiles_max), dim3(256), 0, stream,
                       ws_norm, We, ws_rt, d_rnan);

    // 3) gather-dot per edge + labels (one wave per edge, 8 waves / block)
    const int etot    = epos + eneg;
    const int eblocks = (etot + 7) / 8;
    hipLaunchKernelGGL(hg_edges, dim3(eblocks), dim3(256), 0, stream,
                       ws_rt, ws_norm, d_rnan,
                       pos_rna, pos_dis, neg_rna, neg_dis,
                       out, epos, eneg);
}